// overAll_37606733644143
// MI455X (gfx1250) — compile-verified
//
#include <hip/hip_runtime.h>
#include <hip/hip_bf16.h>
#include <math.h>

// ---------------------------------------------------------------------------
// Problem constants (match reference)
// ---------------------------------------------------------------------------
#define NN   100000   // nodes
#define RR   2048     // relations
#define DD   128      // feature dim
#define EE   500000   // edges
#define ERN  750000   // g_r edges
#define EAN  600000   // ent adjacency entries
#define RAN  600000   // rel adjacency entries
#define PP   64       // proxies
#define F3N  384      // D*(DEPTH+1)
#define EPSF 1e-12f

typedef __attribute__((ext_vector_type(2))) float v2f;
typedef __attribute__((ext_vector_type(8))) float v8f;

// ---------------------------------------------------------------------------
// Small helpers
// ---------------------------------------------------------------------------
__device__ __forceinline__ float waveRedSum(float v) {
#pragma unroll
  for (int o = 16; o > 0; o >>= 1) v += __shfl_xor(v, o, 32);
  return v;
}
__device__ __forceinline__ float waveRedMax(float v) {
#pragma unroll
  for (int o = 16; o > 0; o >>= 1) v = fmaxf(v, __shfl_xor(v, o, 32));
  return v;
}
// float atomic max via signed-max / unsigned-min bit ordering.
// Works with buffer pre-filled with 0xFF800000 (-inf).
__device__ __forceinline__ void atomicMaxF(float* addr, float v) {
  if (v >= 0.0f) atomicMax((int*)addr, __float_as_int(v));
  else           atomicMin((unsigned int*)addr, __float_as_uint(v));
}

__global__ void fill_u32_kernel(unsigned int* p, unsigned int v, int n) {
  int t = blockIdx.x * blockDim.x + threadIdx.x;
  if (t < n) p[t] = v;
}

// ---------------------------------------------------------------------------
// Segment mean scatter: one wave (32 lanes) per entry, float4 per lane.
// acc[row[e], :] += emb[col[e], :], cnt[row[e]] += 1
// ---------------------------------------------------------------------------
__global__ void scatter_add_emb_kernel(const float4* __restrict__ emb,
                                       const int* __restrict__ col,
                                       const int* __restrict__ row,
                                       float* __restrict__ acc,
                                       float* __restrict__ cnt, int n) {
  int t = blockIdx.x * blockDim.x + threadIdx.x;
  int e = t >> 5;
  if (e >= n) return;
  int lane = t & 31;
  int r = row[e], c = col[e];
  float4 v = emb[(size_t)c * 32 + lane];
  float* dst = acc + (size_t)r * DD + lane * 4;
  atomicAdd(dst + 0, v.x);
  atomicAdd(dst + 1, v.y);
  atomicAdd(dst + 2, v.z);
  atomicAdd(dst + 3, v.w);
  if (lane == 0) atomicAdd(cnt + r, 1.0f);
}

// acc[n,d] /= max(cnt[n],1)
__global__ void seg_div_kernel(float* __restrict__ acc,
                               const float* __restrict__ cnt, int total) {
  int t = blockIdx.x * blockDim.x + threadIdx.x;
  if (t >= total) return;
  float c = fmaxf(cnt[t >> 7], 1.0f);  // 128 cols per row
  acc[t] /= c;
}

// rels: divide by count then L2-normalize each 128-row. One wave per edge.
__global__ void rels_finalize_kernel(float4* __restrict__ rels,
                                     const float* __restrict__ cnt, int n) {
  int t = blockIdx.x * blockDim.x + threadIdx.x;
  int e = t >> 5;
  if (e >= n) return;
  int lane = t & 31;
  float4* p = rels + (size_t)e * 32 + lane;
  float4 v = *p;
  float c = fmaxf(cnt[e], 1.0f);
  v.x /= c; v.y /= c; v.z /= c; v.w /= c;
  float ss = v.x * v.x + v.y * v.y + v.z * v.z + v.w * v.w;
  ss = waveRedSum(ss);
  float inv = 1.0f / fmaxf(sqrtf(ss), EPSF);
  v.x *= inv; v.y *= inv; v.z *= inv; v.w *= inv;
  *p = v;
}

// dst[r, c] = tanh(src[r*128 + c]) with dst row stride 384 (column block)
__global__ void tanh_copy_kernel(const float* __restrict__ src,
                                 float* __restrict__ dst, int total) {
  int t = blockIdx.x * blockDim.x + threadIdx.x;
  if (t >= total) return;
  int r = t >> 7, c = t & 127;
  dst[(size_t)r * F3N + c] = tanhf(src[t]);
}

// att1[e] = dot(rels[e,:], ak); atomicMax into segmax[dst[e]]. Wave per edge.
__global__ void att_logits_kernel(const float4* __restrict__ rels,
                                  const float4* __restrict__ ak,
                                  const int* __restrict__ dst,
                                  float* __restrict__ attb,
                                  float* __restrict__ segmax, int n) {
  int t = blockIdx.x * blockDim.x + threadIdx.x;
  int e = t >> 5;
  if (e >= n) return;
  int lane = t & 31;
  float4 r = rels[(size_t)e * 32 + lane];
  float4 a = ak[lane];
  float p = r.x * a.x + r.y * a.y + r.z * a.z + r.w * a.w;
  p = waveRedSum(p);
  if (lane == 0) {
    attb[e] = p;
    atomicMaxF(segmax + dst[e], p);
  }
}

// attb[e] = exp(attb[e] - segmax[dst[e]]); atomicAdd into segsum[dst[e]]
__global__ void att_exp_kernel(const int* __restrict__ dst,
                               float* __restrict__ attb,
                               const float* __restrict__ segmax,
                               float* __restrict__ segsum, int n) {
  int e = blockIdx.x * blockDim.x + threadIdx.x;
  if (e >= n) return;
  int d0 = dst[e];
  float x = expf(attb[e] - segmax[d0]);
  attb[e] = x;
  atomicAdd(segsum + d0, x);
}

// neighs = feats[src]; reflect against rels; scale by softmax weight;
// atomicAdd into acc[dst]. One wave per edge, float4 per lane.
__global__ void edge_scatter_kernel(const float* __restrict__ prev, // outb col block base
                                    const float4* __restrict__ rels,
                                    const int* __restrict__ src,
                                    const int* __restrict__ dst,
                                    const float* __restrict__ attb,
                                    const float* __restrict__ segsum,
                                    float* __restrict__ acc, int n) {
  int t = blockIdx.x * blockDim.x + threadIdx.x;
  int e = t >> 5;
  if (e >= n) return;
  int lane = t & 31;
  int s = src[e], d0 = dst[e];
  float w = attb[e] / segsum[d0];
  const float4 nv = *((const float4*)(prev + (size_t)s * F3N) + lane);
  const float4 rv = rels[(size_t)e * 32 + lane];
  float dp = nv.x * rv.x + nv.y * rv.y + nv.z * rv.z + nv.w * rv.w;
  dp = waveRedSum(dp);
  float c = -2.0f * dp;
  float* ap = acc + (size_t)d0 * DD + lane * 4;
  atomicAdd(ap + 0, (nv.x + c * rv.x) * w);
  atomicAdd(ap + 1, (nv.y + c * rv.y) * w);
  atomicAdd(ap + 2, (nv.z + c * rv.z) * w);
  atomicAdd(ap + 3, (nv.w + c * rv.w) * w);
}

// Row L2-normalize (width 384). One wave per row.
__global__ void l2_rows_kernel(const float* __restrict__ src,
                               float* __restrict__ dst, int n) {
  int t = blockIdx.x * blockDim.x + threadIdx.x;
  int r = t >> 5;
  if (r >= n) return;
  int lane = t & 31;
  const float* p = src + (size_t)r * F3N;
  float ss = 0.0f;
  for (int j = lane; j < F3N; j += 32) { float v = p[j]; ss += v * v; }
  ss = waveRedSum(ss);
  float inv = 1.0f / fmaxf(sqrtf(ss), EPSF);
  float* q = dst + (size_t)r * F3N;
  for (int j = lane; j < F3N; j += 32) q[j] = p[j] * inv;
}

// L2-normalize proxy rows [64,384] and store transposed [384,64].
__global__ void proxy_norm_t_kernel(const float* __restrict__ proxy,
                                    float* __restrict__ outT) {
  int t = blockIdx.x * blockDim.x + threadIdx.x;
  int r = t >> 5;
  if (r >= PP) return;
  int lane = t & 31;
  const float* p = proxy + (size_t)r * F3N;
  float ss = 0.0f;
  for (int j = lane; j < F3N; j += 32) { float v = p[j]; ss += v * v; }
  ss = waveRedSum(ss);
  float inv = 1.0f / fmaxf(sqrtf(ss), EPSF);
  for (int j = lane; j < F3N; j += 32) outT[(size_t)j * PP + r] = p[j] * inv;
}

// Row softmax over 64 columns in place. One wave per row, 2 cols/lane.
__global__ void softmax64_kernel(float* __restrict__ x, int n) {
  int t = blockIdx.x * blockDim.x + threadIdx.x;
  int r = t >> 5;
  if (r >= n) return;
  int lane = t & 31;
  float* p = x + (size_t)r * PP;
  float a0 = p[lane], a1 = p[lane + 32];
  float m = waveRedMax(fmaxf(a0, a1));
  float e0 = expf(a0 - m), e1 = expf(a1 - m);
  float s = waveRedSum(e0 + e1);
  p[lane] = e0 / s;
  p[lane + 32] = e1 / s;
}

// b[i] = a[i] - b[i]   (pf = out - pa@proxy)
__global__ void sub_into_kernel(const float* __restrict__ a,
                                float* __restrict__ b, int total) {
  int t = blockIdx.x * blockDim.x + threadIdx.x;
  if (t >= total) return;
  b[t] = a[t] - b[t];
}

// gate = sigmoid(lin + bias); d_out[:, colOff:colOff+384] = g*out + (1-g)*pf
__global__ void final_gate_kernel(const float* __restrict__ lin,
                                  const float* __restrict__ bias,
                                  const float* __restrict__ outb,
                                  const float* __restrict__ pf,
                                  float* __restrict__ dout,
                                  int total, int colOff) {
  int t = blockIdx.x * blockDim.x + threadIdx.x;
  if (t >= total) return;
  int r = t / F3N, j = t - r * F3N;
  float g = 1.0f / (1.0f + expf(-(lin[t] + bias[j])));
  dout[(size_t)r * (2 * F3N) + colOff + j] = g * outb[t] + (1.0f - g) * pf[t];
}

// ---------------------------------------------------------------------------
// Generic fp32 WMMA GEMM: C[M,Nt] = A[M,K] @ B[K,Nt], all row-major.
// One wave computes a 16x64 C tile: 4 accumulators share one A fragment, so
// each K-step issues 4x V_WMMA_F32_16X16X4_F32 per A load (4x A reuse).
// Fragment layouts per CDNA5 ISA sec. 7.12.2 (32-bit A 16x4, 32-bit C 16x16).
// Requires: M % 16 == 0, Nt % 64 == 0, K % 4 == 0. EXEC stays all-ones in
// active waves (tile-bound branch is wave-uniform).
// ---------------------------------------------------------------------------
__global__ void wmma_gemm_f32_kernel(const float* __restrict__ A,
                                     const float* __restrict__ B,
                                     float* __restrict__ C,
                                     int M, int Nt, int K,
                                     int lda, int ldb, int ldc) {
  int gw = (blockIdx.x * blockDim.x + threadIdx.x) >> 5;
  int tilesN = Nt >> 6;   // 64-wide wave tiles
  int tilesM = M >> 4;
  if (gw >= tilesM * tilesN) return;
  int tm = gw / tilesN;
  int tn = gw - tm * tilesN;

  int lane = threadIdx.x & 31;
  int half = lane >> 4;   // 0: K+{0,1}/M rows 0-7 ; 1: K+{2,3}/M rows 8-15
  int l    = lane & 15;

  // A fragment: lane(l,half) holds A[tm*16+l][k + 2*half + {0,1}]
  const float* Ap = A + (size_t)(tm * 16 + l) * lda + 2 * half;
  // B fragments: lane(l,half) holds B[k + 2*half + {0,1}][tn*64 + j*16 + l]
  const float* Bp = B + (size_t)(2 * half) * ldb + tn * 64 + l;

  v8f acc0 = {}, acc1 = {}, acc2 = {}, acc3 = {};
#pragma unroll 2
  for (int k = 0; k < K; k += 4) {
    v2f a;
    a.x = Ap[k];
    a.y = Ap[k + 1];
    const float* bk = Bp + (size_t)k * ldb;
    v2f b0, b1, b2, b3;
    b0.x = bk[0];  b0.y = bk[ldb + 0];
    b1.x = bk[16]; b1.y = bk[ldb + 16];
    b2.x = bk[32]; b2.y = bk[ldb + 32];
    b3.x = bk[48]; b3.y = bk[ldb + 48];
    acc0 = __builtin_amdgcn_wmma_f32_16x16x4_f32(false, a, false, b0,
                                                 (short)0, acc0, false, false);
    acc1 = __builtin_amdgcn_wmma_f32_16x16x4_f32(false, a, false, b1,
                                                 (short)0, acc1, false, false);
    acc2 = __builtin_amdgcn_wmma_f32_16x16x4_f32(false, a, false, b2,
                                                 (short)0, acc2, false, false);
    acc3 = __builtin_amdgcn_wmma_f32_16x16x4_f32(false, a, false, b3,
                                                 (short)0, acc3, false, false);
  }

  // C store: VGPR i -> row tm*16 + half*8 + i, col tn*64 + j*16 + l
  float* Cp = C + (size_t)(tm * 16 + half * 8) * ldc + tn * 64 + l;
#pragma unroll
  for (int i = 0; i < 8; ++i) {
    size_t ro = (size_t)i * ldc;
    Cp[ro + 0]  = acc0[i];
    Cp[ro + 16] = acc1[i];
    Cp[ro + 32] = acc2[i];
    Cp[ro + 48] = acc3[i];
  }
}

// ---------------------------------------------------------------------------
// Host driver
// ---------------------------------------------------------------------------
static inline int nblk(long long threads, int bs) {
  return (int)((threads + bs - 1) / bs);
}

extern "C" void kernel_launch(void* const* d_in, const int* in_sizes, int n_in,
                              void* d_out, int out_size, void* d_ws, size_t ws_size,
                              hipStream_t stream) {
  (void)in_sizes; (void)n_in; (void)out_size; (void)ws_size;
  const float* ent_emb = (const float*)d_in[0];
  const float* rel_emb = (const float*)d_in[1];
  const float* e_gate  = (const float*)d_in[2];
  const float* e_proxy = (const float*)d_in[3];
  const float* e_bias  = (const float*)d_in[4];
  const float* e_ak0   = (const float*)d_in[5];
  const float* e_ak1   = (const float*)d_in[6];
  const float* r_gate  = (const float*)d_in[7];
  const float* r_proxy = (const float*)d_in[8];
  const float* r_bias  = (const float*)d_in[9];
  const float* r_ak0   = (const float*)d_in[10];
  const float* r_ak1   = (const float*)d_in[11];
  const int* src    = (const int*)d_in[12];
  const int* dst    = (const int*)d_in[13];
  const int* gr_rel = (const int*)d_in[14];
  const int* gr_idx = (const int*)d_in[15];
  const int* ent_row = (const int*)d_in[16];
  const int* ent_col = (const int*)d_in[17];
  const int* rel_row = (const int*)d_in[18];
  const int* rel_col = (const int*)d_in[19];
  float* dout = (float*)d_out;

  // Workspace layout (floats)
  float* W = (float*)d_ws;
  size_t o = 0;
  float* rels    = W + o; o += (size_t)EE * DD;   // [E,128] normalized rel means
  float* attb    = W + o; o += EE;                // per-edge logits / exp
  float* grcnt   = W + o; o += EE;                // rels counts
  float* cnt     = W + o; o += NN;                // seg-mean counts
  float* segmax  = W + o; o += NN;
  float* segsum  = W + o; o += NN;
  float* feat    = W + o; o += (size_t)NN * DD;   // feature / layer accumulator
  float* outb    = W + o; o += (size_t)NN * F3N;  // concat(outs) per tower
  float* B1      = W + o; o += (size_t)NN * F3N;  // l2n(out) then gate-linear
  float* B2      = W + o; o += (size_t)NN * F3N;  // pa@proxy then pf
  float* pabuf   = W + o; o += (size_t)NN * PP;   // logits / pa
  float* nproxyT = W + o; o += (size_t)F3N * PP;  // l2n(proxy)^T

  const int BS = 256;

  // ---- rels = l2n(seg_mean(rel_emb[gr_rel], gr_idx, E)) -------------------
  hipMemsetAsync(rels, 0, (size_t)EE * DD * sizeof(float), stream);
  hipMemsetAsync(grcnt, 0, (size_t)EE * sizeof(float), stream);
  scatter_add_emb_kernel<<<nblk((long long)ERN * 32, BS), BS, 0, stream>>>(
      (const float4*)rel_emb, gr_rel, gr_idx, rels, grcnt, ERN);
  rels_finalize_kernel<<<nblk((long long)EE * 32, BS), BS, 0, stream>>>(
      (float4*)rels, grcnt, EE);

  // ---- two dual towers ----------------------------------------------------
  for (int dual = 0; dual < 2; ++dual) {
    const float* emb  = dual ? rel_emb : ent_emb;
    const int*   arow = dual ? rel_row : ent_row;
    const int*   acol = dual ? rel_col : ent_col;
    const int    nadj = dual ? RAN : EAN;
    const float* gateK = dual ? r_gate  : e_gate;
    const float* proxy = dual ? r_proxy : e_proxy;
    const float* bias  = dual ? r_bias  : e_bias;
    const float* aks[2] = { dual ? r_ak0 : e_ak0, dual ? r_ak1 : e_ak1 };

    // feature = seg_mean(emb[acol], arow, N)
    hipMemsetAsync(feat, 0, (size_t)NN * DD * sizeof(float), stream);
    hipMemsetAsync(cnt, 0, (size_t)NN * sizeof(float), stream);
    scatter_add_emb_kernel<<<nblk((long long)nadj * 32, BS), BS, 0, stream>>>(
        (const float4*)emb, acol, arow, feat, cnt, nadj);
    seg_div_kernel<<<nblk((long long)NN * DD, BS), BS, 0, stream>>>(
        feat, cnt, NN * DD);

    // outs[0] = tanh(feature) -> outb[:, 0:128]
    tanh_copy_kernel<<<nblk((long long)NN * DD, BS), BS, 0, stream>>>(
        feat, outb, NN * DD);

    // attention layers
    for (int layer = 0; layer < 2; ++layer) {
      fill_u32_kernel<<<nblk(NN, BS), BS, 0, stream>>>(
          (unsigned int*)segmax, 0xFF800000u, NN);  // -inf
      hipMemsetAsync(segsum, 0, (size_t)NN * sizeof(float), stream);
      att_logits_kernel<<<nblk((long long)EE * 32, BS), BS, 0, stream>>>(
          (const float4*)rels, (const float4*)aks[layer], dst, attb, segmax, EE);
      att_exp_kernel<<<nblk(EE, BS), BS, 0, stream>>>(
          dst, attb, segmax, segsum, EE);
      hipMemsetAsync(feat, 0, (size_t)NN * DD * sizeof(float), stream);
      edge_scatter_kernel<<<nblk((long long)EE * 32, BS), BS, 0, stream>>>(
          outb + (size_t)layer * DD, (const float4*)rels, src, dst,
          attb, segsum, feat, EE);
      tanh_copy_kernel<<<nblk((long long)NN * DD, BS), BS, 0, stream>>>(
          feat, outb + (size_t)(layer + 1) * DD, NN * DD);
    }

    // B1 = l2n(outb)
    l2_rows_kernel<<<nblk((long long)NN * 32, BS), BS, 0, stream>>>(outb, B1, NN);
    // nproxyT = l2n(proxy)^T  [384,64]
    proxy_norm_t_kernel<<<nblk((long long)PP * 32, BS), BS, 0, stream>>>(
        proxy, nproxyT);

    // logits = B1 @ nproxyT  -> pabuf [N,64]   (WMMA fp32, 16x64 wave tiles)
    {
      long long tiles = (long long)(NN / 16) * (PP / 64);
      wmma_gemm_f32_kernel<<<nblk(tiles * 32, BS), BS, 0, stream>>>(
          B1, nproxyT, pabuf, NN, PP, F3N, F3N, PP, PP);
    }
    softmax64_kernel<<<nblk((long long)NN * 32, BS), BS, 0, stream>>>(pabuf, NN);

    // B2 = pa @ proxy  [N,384]   (WMMA fp32)
    {
      long long tiles = (long long)(NN / 16) * (F3N / 64);
      wmma_gemm_f32_kernel<<<nblk(tiles * 32, BS), BS, 0, stream>>>(
          pabuf, proxy, B2, NN, F3N, PP, PP, F3N, F3N);
    }
    // pf: B2 = outb - B2
    sub_into_kernel<<<nblk((long long)NN * F3N, BS), BS, 0, stream>>>(
        outb, B2, NN * F3N);

    // B1 = pf @ gateK  [N,384]   (WMMA fp32)
    {
      long long tiles = (long long)(NN / 16) * (F3N / 64);
      wmma_gemm_f32_kernel<<<nblk(tiles * 32, BS), BS, 0, stream>>>(
          B2, gateK, B1, NN, F3N, F3N, F3N, F3N, F3N);
    }

    // d_out[:, dual*384 : +384] = g*out + (1-g)*pf
    final_gate_kernel<<<nblk((long long)NN * F3N, BS), BS, 0, stream>>>(
        B1, bias, outb, B2, dout, NN * F3N, dual * F3N);
  }
}